// PredicateCRFLoss_13855564497397
// MI455X (gfx1250) — compile-verified
//
#include <hip/hip_runtime.h>

#define Bn 2048
#define Tn 1024
#define Kn 8
#define WAVES 8
#define LSTR 10              // LDS row stride (even -> aligned b64 loads, conflict-free)

typedef __attribute__((ext_vector_type(2))) float v2f;
typedef __attribute__((ext_vector_type(8))) float v8f;

// ---------------------------------------------------------------------------
// Pass 1: chunked CRF scan. Each wave carries TWO 8x8 log-semiring chunk
// matrices as one 16x8 tile, stored in LINEAR domain:
//     Q[r,j] = exp(X[r,j] - o_r),   o_r tracked in a register at lane r.
// Per step:  A = Q / rowmax(Q);  d = A x exp(trans)  (two chained
// V_WMMA_F32_16X16X4_F32, K=8=2*4);  Q <- d * exp(emit_t);  o_r += log(rowmax)
// (masked: identity when t >= seq_l[b]).  3 transcendentals/lane/step.
// Emission loads are software-pipelined one step ahead so no s_wait_loadcnt
// sits in the serial recurrence.
// ---------------------------------------------------------------------------
__global__ __launch_bounds__(256) void crf_scan_wmma(
    const float* __restrict__ score, const float* __restrict__ trans,
    const int* __restrict__ seq_l, float* __restrict__ Mout, int C, int S)
{
    __shared__ float lds[WAVES][16 * LSTR];
    const int tid  = threadIdx.x;
    const int wid  = tid >> 5;
    const int lane = tid & 31;
    const int g    = lane >> 4;          // K-half for A-layout / row-half for D
    const int r    = lane & 15;          // A-layout row
    const int n    = r;                  // D-layout column
    float* wl = lds[wid];

    const int waveIdx = blockIdx.x * WAVES + wid;

    // D-epilogue half: rows m = v + 8g belong to pair (2w + g).
    const int pairD = waveIdx * 2 + g;
    const int bD = pairD / C, cD = pairD - bD * C;
    const int LD = seq_l[bD];
    const long long emitBase = (long long)bD * Tn * Kn + (n & 7);

    // A-side row r belongs to pair (2w + (r>>3)) -> predicate for o_r update.
    const int pairA = waveIdx * 2 + (r >> 3);
    const int bA = pairA / C, cA = pairA - bA * C;
    const int LA = seq_l[bA];

    // B-matrix = exp(transitions), constant. 32-bit B 4x16 layout (mirror of A):
    // slot s at lane -> (K = 2g+s, N = n).
    v2f Bm0, Bm1;
    {
        float t00 = (n < 8) ? __expf(trans[(2 * g + 0) * Kn + (n & 7)]) : 0.f;
        float t01 = (n < 8) ? __expf(trans[(2 * g + 1) * Kn + (n & 7)]) : 0.f;
        float t10 = (n < 8) ? __expf(trans[(4 + 2 * g + 0) * Kn + (n & 7)]) : 0.f;
        float t11 = (n < 8) ? __expf(trans[(4 + 2 * g + 1) * Kn + (n & 7)]) : 0.f;
        Bm0 = v2f{t00, t01};
        Bm1 = v2f{t10, t11};
    }

    // Linear-domain log-semiring identity: Q = I, o = 0.
    if (lane < 16) {
        #pragma unroll
        for (int k = 0; k < 8; ++k)
            wl[r * LSTR + k] = (k == (r & 7)) ? 1.f : 0.f;
    }
    float o = 0.f;                       // log offset of row r
    // LDS is in-order within a wave; each wave owns its slice -> no barrier.

    const int i0 = 2 * g;                // A slots: i0, i0+1, i0+4, i0+5 (even -> b64)

    // Prime the emission pipeline with step tl = 1 (clamped, always in-bounds).
    {
        const int t1 = cD * S + 1;
        const int tc1 = (t1 < Tn) ? t1 : (Tn - 1);
        // first value loaded below inside the loop preamble pattern
        (void)tc1;
    }
    int tCur = cD * S + 1;
    int tcCur = (tCur < Tn) ? tCur : (Tn - 1);
    float emitNext = score[emitBase + (long long)tcCur * Kn];

    for (int tl = 1; tl <= S; ++tl) {
        // Issue step (tl+1)'s emission load now; consume the value loaded for tl.
        const float emit = emitNext;
        const int tN  = cD * S + tl + 1;
        const int tcN = (tN < Tn) ? tN : (Tn - 1);
        emitNext = score[emitBase + (long long)tcN * Kn];
        const int tp = (tcN + 4 < Tn) ? tcN + 4 : tcN;
        __builtin_prefetch(&score[emitBase + (long long)tp * Kn], 0, 1);

        float2 qa = *reinterpret_cast<float2*>(wl + r * LSTR + i0);
        float2 qb = *reinterpret_cast<float2*>(wl + r * LSTR + i0 + 4);
        float qp   = fmaxf(fmaxf(qa.x, qa.y), fmaxf(qb.x, qb.y));
        float qmax = fmaxf(qp, __shfl_xor(qp, 16, 32));      // full row max
        float rinv = __builtin_amdgcn_rcpf(qmax);

        v2f A0 = v2f{qa.x * rinv, qa.y * rinv};
        v2f A1 = v2f{qb.x * rinv, qb.y * rinv};

        union { v8f v; float f[8]; } du;
        v8f zero = {0.f, 0.f, 0.f, 0.f, 0.f, 0.f, 0.f, 0.f};
        du.v = __builtin_amdgcn_wmma_f32_16x16x4_f32(false, A0, false, Bm0,
                                                     (short)0, zero, false, false);
        du.v = __builtin_amdgcn_wmma_f32_16x16x4_f32(false, A1, false, Bm1,
                                                     (short)0, du.v, false, false);

        const float ee = __expf(emit);               // emit loaded last iteration

        // Offset update for row r (predicate of row r's own pair); no broadcast:
        // qmax_r is already resident at lanes r and r+16.
        const int tA = cA * S + tl;
        float lg = __logf(qmax);
        o += (tA < LA) ? lg : 0.f;

        // Single predicated store region (masked step == identity).
        const int tD = cD * S + tl;
        if ((tD < LD) && (n < 8)) {
            #pragma unroll
            for (int v = 0; v < 8; ++v)
                wl[(v + 8 * g) * LSTR + n] = du.f[v] * ee;
        }
    }

    // Emit chunk matrices in log domain: X = log(Q) + o_r.
    // Lane writes row r, cols [4g, 4g+4); o is exactly row r's offset.
    long long base = (long long)pairA * 64 + (long long)(r & 7) * 8 + 4 * g;
    #pragma unroll
    for (int q = 0; q < 4; ++q)
        Mout[base + q] = __logf(wl[r * LSTR + 4 * g + q]) + o;
}

// ---------------------------------------------------------------------------
// Pass 2: compose the C chunk matrices per batch row, fold alpha0 and end
// transitions -> log Z_b.  (M entries may be -inf from log(0): exp(-inf)=0.)
// ---------------------------------------------------------------------------
__global__ __launch_bounds__(256) void crf_compose(
    const float* __restrict__ score, const float* __restrict__ startT,
    const float* __restrict__ endT, const float* __restrict__ M,
    float* __restrict__ logz, int C)
{
    int b = blockIdx.x * blockDim.x + threadIdx.x;
    if (b >= Bn) return;
    float al[8];
    #pragma unroll
    for (int j = 0; j < 8; ++j)
        al[j] = startT[j] + score[(long long)b * Tn * Kn + j];

    for (int c = 0; c < C; ++c) {
        const float* Mb = M + ((long long)b * C + c) * 64;
        float Ml[64];
        #pragma unroll
        for (int e = 0; e < 64; ++e) Ml[e] = Mb[e];
        float na[8];
        #pragma unroll
        for (int j = 0; j < 8; ++j) {
            float mx = -3.0e38f;
            #pragma unroll
            for (int k = 0; k < 8; ++k) mx = fmaxf(mx, al[k] + Ml[k * 8 + j]);
            float s = 0.f;
            #pragma unroll
            for (int k = 0; k < 8; ++k) s += __expf(al[k] + Ml[k * 8 + j] - mx);
            na[j] = __logf(s) + mx;
        }
        #pragma unroll
        for (int j = 0; j < 8; ++j) al[j] = na[j];
    }
    float mx = -3.0e38f;
    #pragma unroll
    for (int j = 0; j < 8; ++j) mx = fmaxf(mx, al[j] + endT[j]);
    float s = 0.f;
    #pragma unroll
    for (int j = 0; j < 8; ++j) s += __expf(al[j] + endT[j] - mx);
    logz[b] = __logf(s) + mx;
}

// ---------------------------------------------------------------------------
// Pass 3: gold-path numerator, one wave per batch row; val[b] = logZ_b - num_b.
// ---------------------------------------------------------------------------
__global__ __launch_bounds__(256) void crf_num(
    const float* __restrict__ score, const float* __restrict__ trans,
    const float* __restrict__ startT, const float* __restrict__ endT,
    const int* __restrict__ tags, const int* __restrict__ seq_l,
    const float* __restrict__ logz, float* __restrict__ val)
{
    int wid = threadIdx.x >> 5, lane = threadIdx.x & 31;
    int b = blockIdx.x * (blockDim.x >> 5) + wid;
    if (b >= Bn) return;
    int L = seq_l[b];
    const int*   tg = tags  + (long long)b * Tn;
    const float* sc = score + (long long)b * Tn * Kn;
    float p = 0.f;
    for (int t = lane; t < Tn; t += 32) {
        if (t < L) {
            int c = tg[t];
            p += sc[t * Kn + c];
            if (t >= 1) p += trans[tg[t - 1] * Kn + c];
        }
    }
    #pragma unroll
    for (int off = 16; off; off >>= 1) p += __shfl_xor(p, off, 32);
    if (lane == 0) {
        p += startT[tg[0]] + endT[tg[L - 1]];
        val[b] = logz[b] - p;
    }
}

// Pass 4: deterministic fixed-order reduction -> d_out[0] = sum(val)/B.
__global__ void crf_reduce(const float* __restrict__ val, float* __restrict__ out)
{
    __shared__ float sh[256];
    int tid = threadIdx.x;
    float s = 0.f;
    #pragma unroll
    for (int q = 0; q < 8; ++q) s += val[tid * 8 + q];
    sh[tid] = s;
    __syncthreads();
    for (int o = 128; o; o >>= 1) {
        if (tid < o) sh[tid] += sh[tid + o];
        __syncthreads();
    }
    if (tid == 0) out[0] = sh[0] / (float)Bn;
}

// Pass 5: masked argmax over K=8 (first-max tie-break, like jnp.argmax).
__global__ __launch_bounds__(256) void crf_pred(
    const float* __restrict__ logp, const int* __restrict__ seq_l,
    float* __restrict__ out)
{
    long long i = (long long)blockIdx.x * blockDim.x + threadIdx.x;
    if (i >= (long long)Bn * Tn) return;
    int b = (int)(i >> 10);
    int t = (int)(i & (Tn - 1));
    const float4* p = (const float4*)(logp + i * Kn);
    float4 x0 = p[0], x1 = p[1];
    float v[8] = {x0.x, x0.y, x0.z, x0.w, x1.x, x1.y, x1.z, x1.w};
    float best = v[0];
    int arg = 0;
    #pragma unroll
    for (int j = 1; j < 8; ++j)
        if (v[j] > best) { best = v[j]; arg = j; }
    out[i] = (t < seq_l[b]) ? (float)arg : 0.f;
}

extern "C" void kernel_launch(void* const* d_in, const int* in_sizes, int n_in,
                              void* d_out, int out_size, void* d_ws, size_t ws_size,
                              hipStream_t stream)
{
    (void)in_sizes; (void)n_in; (void)out_size;
    const float* log_p  = (const float*)d_in[0];
    const float* score  = (const float*)d_in[1];
    const float* trans  = (const float*)d_in[2];
    const float* startT = (const float*)d_in[3];
    const float* endT   = (const float*)d_in[4];
    const int*   tags   = (const int*)d_in[5];
    const int*   seq_l  = (const int*)d_in[6];
    float* out = (float*)d_out;

    // Pick chunk count C from workspace size (deterministic per call).
    int C = 8;
    while (C > 1 && ws_size < ((size_t)Bn * C * 64 + 2 * Bn) * sizeof(float))
        C >>= 1;
    const int S = Tn / C;

    float* Mws  = (float*)d_ws;                       // B*C 8x8 chunk matrices
    float* logz = Mws + (size_t)Bn * C * 64;          // B floats
    float* val  = logz + Bn;                          // B floats

    crf_scan_wmma<<<(Bn * C) / 16, 256, 0, stream>>>(score, trans, seq_l, Mws, C, S);
    crf_compose<<<Bn / 256, 256, 0, stream>>>(score, startT, endT, Mws, logz, C);
    crf_num<<<Bn / 8, 256, 0, stream>>>(score, trans, startT, endT, tags, seq_l, logz, val);
    crf_reduce<<<1, 256, 0, stream>>>(val, out);
    crf_pred<<<(Bn * Tn) / 256, 256, 0, stream>>>(log_p, seq_l, out + 1);
}